// Decoder_10866267258962
// MI455X (gfx1250) — compile-verified
//
#include <hip/hip_runtime.h>
#include <hip/hip_bf16.h>

#define NTOT   131072
#define NSIG   64
#define SEG    2048
#define NHID   256

typedef __attribute__((ext_vector_type(2))) float v2f;
typedef __attribute__((ext_vector_type(8))) float v8f;
typedef unsigned int u32x4 __attribute__((ext_vector_type(4)));
typedef int          i32x8 __attribute__((ext_vector_type(8)));
typedef int          i32x4 __attribute__((ext_vector_type(4)));

// ---------------- Kernel 1: sigma-predictor MLP (one workgroup) ----------------
__global__ __launch_bounds__(256) void mlp_sigmas_kernel(
    const float* __restrict__ x, const float* __restrict__ W1,
    const float* __restrict__ b1, const float* __restrict__ W2,
    const float* __restrict__ b2, float* __restrict__ sigmas) {
  __shared__ float s_x[NHID];
  __shared__ float s_s[NHID];
  const int t = threadIdx.x;
  s_x[t] = x[t];
  __syncthreads();
  float acc = b1[t];
#pragma unroll 8
  for (int i = 0; i < NHID; ++i) acc += s_x[i] * W1[i * NHID + t];
  s_s[t] = acc / (1.0f + __expf(-acc));            // swish = x*sigmoid(x)
  __syncthreads();
  if (t < NSIG) {
    float a2 = b2[t];
#pragma unroll 8
    for (int h = 0; h < NHID; ++h) a2 += s_s[h] * W2[h * NSIG + t];
    sigmas[t] = a2;
  }
}

// ---------------- Kernel 2: normalized Gaussian windows (one block / sigma) ----
__global__ __launch_bounds__(256) void windows_kernel(
    const float* __restrict__ sigmas, float* __restrict__ win) {
  __shared__ float red[256];
  const int s = blockIdx.x, tid = threadIdx.x;
  const float sig = sigmas[s];
  const float inv2s2 = 1.0f / (2.0f * sig * sig);
  float vals[SEG / 256];
  float lsum = 0.0f;
#pragma unroll
  for (int k = 0; k < SEG / 256; ++k) {
    float tf = (float)(k * 256 + tid) - 1024.0f;   // mu = SEG/2
    float v = __expf(-tf * tf * inv2s2);
    vals[k] = v;
    lsum += v;
  }
  red[tid] = lsum;
  __syncthreads();
  for (int off = 128; off > 0; off >>= 1) {
    if (tid < off) red[tid] += red[tid + off];
    __syncthreads();
  }
  const float inv = 1.0f / red[0];
#pragma unroll
  for (int k = 0; k < SEG / 256; ++k)
    win[s * SEG + k * 256 + tid] = vals[k] * inv;
}

// ---------------- Kernel 3: y = x @ W3 + b3 (HBM-bound matvec, b128 loads) ----
__global__ __launch_bounds__(256) void matvec_kernel(
    const float* __restrict__ x, const float* __restrict__ W3,
    const float* __restrict__ b3, float* __restrict__ y) {
  __shared__ float s_x[NHID];
  const int tid = threadIdx.x;
  s_x[tid] = x[tid];
  __syncthreads();
  const int j = (blockIdx.x * 256 + tid) * 4;
  float4 acc = *(const float4*)(b3 + j);
#pragma unroll 4
  for (int i = 0; i < NHID; ++i) {
    const float4 w = *(const float4*)(W3 + (size_t)i * NTOT + j);
    const float xv = s_x[i];
    acc.x += xv * w.x; acc.y += xv * w.y;
    acc.z += xv * w.z; acc.w += xv * w.w;
  }
  *(float4*)(y + j) = acc;
}

// ----- TDM: 1-D contiguous tile (nelem f32) Global -> LDS via D# descriptor ---
#if defined(__HIP_DEVICE_COMPILE__) && __has_builtin(__builtin_amdgcn_tensor_load_to_lds)
#define HAVE_TDM 1
__device__ __forceinline__ void tdm_load_1d(const float* gsrc,
                                            unsigned lds_byte_off, int nelem) {
  const unsigned long long ga = (unsigned long long)(uintptr_t)(const void*)gsrc;
  u32x4 g0;
  g0[0] = 1u;                                           // count=1 (valid), user mode
  g0[1] = lds_byte_off;                                 // lds_addr
  g0[2] = (unsigned)(ga & 0xffffffffu);                 // global_addr[31:0]
  g0[3] = (unsigned)((ga >> 32) & 0x01ffffffu) | (2u << 30);  // addr[56:32], type=2
  i32x8 g1;
  g1[0] = 0x00020000;                 // data_size=2 (4B); no pad/iterate/multicast
  g1[1] = (int)((unsigned)nelem << 16);   // tensor_dim0[15:0]
  g1[2] = 0x00010000;                 // tensor_dim0[31:16]=0, tensor_dim1=1
  g1[3] = (int)((unsigned)nelem << 16);   // tile_dim0
  g1[4] = 1;                          // tile_dim1=1, tile_dim2=0
  g1[5] = nelem;                      // tensor_dim0_stride[31:0]
  g1[6] = 0;
  g1[7] = 0;
  i32x4 gz4 = {0, 0, 0, 0};           // groups 2/3 unused (<=2D tensor)
  i32x8 gz8 = {0, 0, 0, 0, 0, 0, 0, 0};
  // 6-arg form (clang-23 / therock-10.0 headers): groups 0..3 + extra group + cpol
  __builtin_amdgcn_tensor_load_to_lds(g0, g1, gz4, gz4, gz8, 0);
}
#else
#define HAVE_TDM 0
#endif

// ---------------- Kernel 4: per-segment convolution via f32 WMMA --------------
// out[n] = sum_k seg[k] * ww[n-k],  ww[d] = window[d+1023]  (np.convolve 'same')
// Tile: D[m][nn] = out[16*(16*wave+nn)+m]; k = 16*(16*wave+nn) - 4u + kk.
//   A[m][kk]  = ww[m + 4u - kk]                      (Toeplitz window slice)
//   B[kk][nn] = seg[256*wave + 16*nn + kk - 4u]      (contiguous seg tile)
// Serial accumulation over diagonal u; padded LDS images remove range checks.
__global__ __launch_bounds__(256) void conv_wmma_kernel(
    const float* __restrict__ win, const float* __restrict__ y,
    float* __restrict__ out) {
  __shared__ float s_win[2084];   // window[j] at s_win[j + 17], zero pads
  __shared__ float s_seg[4096];   // seg[j]    at s_seg[j + 1024], zero pads
  const int seg = blockIdx.x;
  const int tid = threadIdx.x;
  const int segbase = seg * SEG;

#if HAVE_TDM
  // Zero only the pad halos with DS stores; bulk data arrives via TDM DMA.
  for (int k = tid; k < 2084; k += 256)
    if (k < 17 || k >= 17 + SEG) s_win[k] = 0.0f;
  for (int k = tid; k < 4096; k += 256)
    if (k < 1024 || k >= 1024 + SEG) s_seg[k] = 0.0f;
  if (tid < 32) {                       // wave 0 issues both tensor DMAs
    tdm_load_1d(win + segbase, (unsigned)(uintptr_t)s_win + 17u * 4u, SEG);
    tdm_load_1d(y + segbase, (unsigned)(uintptr_t)s_seg + 1024u * 4u, SEG);
    __builtin_amdgcn_s_wait_tensorcnt(0);
  }
#else
  for (int k = tid; k < 2084; k += 256) {
    const int src = k - 17;
    s_win[k] = (src >= 0 && src < SEG) ? win[segbase + src] : 0.0f;
  }
  for (int k = tid; k < 4096; k += 256) {
    const int src = k - 1024;
    s_seg[k] = (src >= 0 && src < SEG) ? y[segbase + src] : 0.0f;
  }
#endif
  __syncthreads();

  const int wave = tid >> 5;          // 8 waves -> 8 tiles of 256 outputs
  const int lane = tid & 31;
  const int col  = lane & 15;         // A row m == B col nn == D col nn
  const int half = lane >> 4;
  const int kk0  = half * 2;          // VGPR0 holds K=kk0, VGPR1 holds K=kk0+1

  // u-range = (window support) ∩ (segment support); lengths are multiples of 4
  const int ul = 64 * wave - 511;
  const int uh = 64 * wave + 60;
  const int ulo = (ul > -259) ? ul : -259;
  const int uhi = (uh < 256) ? uh : 256;

  int dw = col - kk0 + 1023 + 17 + 4 * ulo;              // a.x; a.y at dw-1
  int ds = 1024 + wave * 256 + col * 16 + kk0 - 4 * ulo; // b.x; b.y at ds+1

  v8f c = {};
  const int nQuad = (uhi - ulo + 1) >> 2;   // trip counts 320..512, all %4==0
  for (int it = 0; it < nQuad; ++it) {
#pragma unroll
    for (int s = 0; s < 4; ++s) {
      v2f a, b;
      a.x = s_win[dw];
      a.y = s_win[dw - 1];
      b.x = s_seg[ds];
      b.y = s_seg[ds + 1];
      c = __builtin_amdgcn_wmma_f32_16x16x4_f32(false, a, false, b, (short)0, c,
                                                false, false);
      dw += 4;
      ds -= 4;
    }
  }

  // D layout: lane holds D[v + 8*half][col] in c[v]
  const int obase = segbase + wave * 256 + col * 16 + half * 8;
#pragma unroll
  for (int v = 0; v < 8; ++v) out[obase + v] = c[v];
}

extern "C" void kernel_launch(void* const* d_in, const int* in_sizes, int n_in,
                              void* d_out, int out_size, void* d_ws, size_t ws_size,
                              hipStream_t stream) {
  const float* x  = (const float*)d_in[0];
  const float* W1 = (const float*)d_in[1];
  const float* b1 = (const float*)d_in[2];
  const float* W2 = (const float*)d_in[3];
  const float* b2 = (const float*)d_in[4];
  const float* W3 = (const float*)d_in[5];
  const float* b3 = (const float*)d_in[6];
  float* out = (float*)d_out;

  float* ws     = (float*)d_ws;
  float* sigmas = ws;                  // 64 floats
  float* win    = ws + 64;             // 64*2048 floats
  float* y      = win + NSIG * SEG;    // 131072 floats  (~1.05 MB total)

  mlp_sigmas_kernel<<<1, 256, 0, stream>>>(x, W1, b1, W2, b2, sigmas);
  windows_kernel<<<NSIG, 256, 0, stream>>>(sigmas, win);
  matvec_kernel<<<NTOT / (256 * 4), 256, 0, stream>>>(x, W3, b3, y);
  conv_wmma_kernel<<<NSIG, 256, 0, stream>>>(win, y, out);
}